// MaxKGIN_62388694942259
// MI455X (gfx1250) — compile-verified
//
#include <hip/hip_runtime.h>
#include <hip/hip_bf16.h>

typedef __attribute__((ext_vector_type(2))) float v2f;
typedef __attribute__((ext_vector_type(8))) float v8f;

#define HID   128
#define KDIM  128
#define MAXKK 16

// ---------------------------------------------------------------------------
// WMMA GEMM: C[M,NCOLS] = op(A[M,128] @ W[128,NCOLS] + bias), op = relu|id
// Block = 256 threads = 8 waves. Wave w computes a 16-row strip across all
// NCOLS columns using v_wmma_f32_16x16x4_f32.
//
// W is staged in LDS *k-pair interleaved*: element (k,n) lives at
//   Ws[((k>>1)*NCOLS + n)*2 + (k&1)]
// so each WMMA B fragment {W[ka][n], W[ka+1][n]} is one aligned ds_load_b64
// directly into an even VGPR pair (no repacking moves).
//
// A fragments are hoisted into registers 16 v2f at a time so global loads
// pipeline instead of serializing each k-step behind s_wait_loadcnt 0.
//
// f32 WMMA fragment layouts (ISA 7.12.2):
//   A 16x4 : lanes 0-15 hold (M=lane, K=k0,k0+1), lanes 16-31 (M=lane-16, K=k0+2,k0+3)
//   B 4x16 : lanes 0-15 hold (K=k0..k0+1, N=lane), lanes 16-31 (K=k0+2..3, N=lane-16)
//   C 16x16: vgpr t: lanes 0-15 -> (M=t, N=lane), lanes 16-31 -> (M=t+8, N=lane-16)
// ---------------------------------------------------------------------------
template<int NCOLS, bool RELU>
__global__ __launch_bounds__(256) void gemm_bias_kernel(
    const float* __restrict__ A, const float* __restrict__ W,
    const float* __restrict__ bias, float* __restrict__ C, int M)
{
    constexpr int CT = NCOLS / 16;
    __shared__ float Ws[KDIM * NCOLS];

    // Cooperative staging of W with k-pair interleave. id -> (n, k-quad).
    {
        constexpr int total4 = KDIM * NCOLS / 4;
        for (int id = threadIdx.x; id < total4; id += 256) {
            const int n  = id & (NCOLS - 1);
            const int kb = (id / NCOLS) * 4;
            v2f w01, w23;
            w01.x = W[(kb + 0) * NCOLS + n];
            w01.y = W[(kb + 1) * NCOLS + n];
            w23.x = W[(kb + 2) * NCOLS + n];
            w23.y = W[(kb + 3) * NCOLS + n];
            *(v2f*)&Ws[(((kb >> 1) + 0) * NCOLS + n) * 2] = w01;
            *(v2f*)&Ws[(((kb >> 1) + 1) * NCOLS + n) * 2] = w23;
        }
    }
    __syncthreads();

    const int wave  = threadIdx.x >> 5;
    const int lane  = threadIdx.x & 31;
    const int strip = blockIdx.x * 8 + wave;
    if (strip * 16 >= M) return;

    const int half = lane >> 4;    // 0: lanes 0-15, 1: lanes 16-31
    const int lrow = lane & 15;
    const float* __restrict__ Arow =
        A + (size_t)(strip * 16 + lrow) * KDIM + 2 * half;

    v8f acc[CT] = {};

#pragma unroll
    for (int kh = 0; kh < 2; ++kh) {
        // Hoist 16 A fragments (one K half) into registers: loads pipeline.
        v2f areg[16];
#pragma unroll
        for (int j = 0; j < 16; ++j)
            areg[j] = *(const v2f*)(Arow + kh * (KDIM / 2) + 4 * j);

#pragma unroll
        for (int kk = 0; kk < KDIM / 2; kk += 4) {
            const int p = ((kh * (KDIM / 2) + kk) >> 1) + half;  // k-pair idx
#pragma unroll
            for (int ct = 0; ct < CT; ++ct) {
                v2f b = *(const v2f*)&Ws[(p * NCOLS + ct * 16 + lrow) * 2];
                acc[ct] = __builtin_amdgcn_wmma_f32_16x16x4_f32(
                    false, areg[kk >> 2], false, b, (short)0, acc[ct],
                    false, false);
            }
        }
    }

#pragma unroll
    for (int ct = 0; ct < CT; ++ct) {
        const int n   = ct * 16 + lrow;
        const float bv = bias[n];
#pragma unroll
        for (int t = 0; t < 8; ++t) {
            float v = acc[ct][t] + bv;
            if (RELU) v = fmaxf(v, 0.0f);
            C[(size_t)(strip * 16 + t + 8 * half) * NCOLS + n] = v;
        }
    }
}

// ---------------------------------------------------------------------------
// MaxK(16 of 128) + eps residual init. One wave per row (8 rows / block).
// Exact rank via count-greater with index tie-break (== jax top_k selection).
// Writes dense Hout = (1+eps)*h_sparse (the aggregation target), and the
// compacted (value, index) lists used for sparse edge propagation. The rank
// of a kept element is a unique slot in [0,16), so no prefix sum is needed.
// ---------------------------------------------------------------------------
__global__ __launch_bounds__(256) void maxk_eps_kernel(
    const float* __restrict__ Z, const float* __restrict__ eps, int layer,
    float* __restrict__ Hout, float* __restrict__ cval, int* __restrict__ cidx,
    int M)
{
    __shared__ float rowbuf[8][HID];
    const int wave = threadIdx.x >> 5;
    const int lane = threadIdx.x & 31;
    const int row  = blockIdx.x * 8 + wave;
    const bool active = row < M;

    float v[4];
    if (active) {
        const float* __restrict__ z = Z + (size_t)row * HID;
#pragma unroll
        for (int j = 0; j < 4; ++j) {
            v[j] = z[lane + 32 * j];
            rowbuf[wave][lane + 32 * j] = v[j];
        }
    }
    __syncthreads();
    if (!active) return;

    const float ep1 = 1.0f + eps[layer];
#pragma unroll
    for (int j = 0; j < 4; ++j) {
        const int   i  = lane + 32 * j;
        const float vi = v[j];
        int rank = 0;
        for (int q = 0; q < HID; ++q) {
            const float vq = rowbuf[wave][q];
            rank += (vq > vi) || (vq == vi && q < i);
        }
        const bool keep = rank < MAXKK;
        Hout[(size_t)row * HID + i] = keep ? ep1 * vi : 0.0f;
        if (keep) {
            cval[(size_t)row * MAXKK + rank] = vi;
            cidx[(size_t)row * MAXKK + rank] = i;
        }
    }
}

// ---------------------------------------------------------------------------
// Sparse GIN aggregation: one lane per (edge, k). h[dst] += h_sparse[src]
// restricted to the 16 kept entries. 16 consecutive lanes share an edge, so
// cval/cidx reads are coalesced; atomics land in the 192MB L2 (h is 51MB).
// ---------------------------------------------------------------------------
__global__ __launch_bounds__(256) void edge_scatter_kernel(
    const int* __restrict__ src, const int* __restrict__ dst,
    const float* __restrict__ cval, const int* __restrict__ cidx,
    float* __restrict__ H, int nEdges)
{
    const long long tid = (long long)blockIdx.x * 256 + threadIdx.x;
    const long long e   = tid >> 4;
    if (e >= nEdges) return;
    const int t = (int)(tid & 15);
    const int u = src[e];
    const int v = dst[e];
    const size_t c = (size_t)u * MAXKK + t;
    atomicAdd(H + (size_t)v * HID + cidx[c], cval[c]);
}

// ---------------------------------------------------------------------------
extern "C" void kernel_launch(void* const* d_in, const int* in_sizes, int n_in,
                              void* d_out, int out_size, void* d_ws, size_t ws_size,
                              hipStream_t stream)
{
    (void)n_in; (void)out_size; (void)ws_size;
    const float* x     = (const float*)d_in[0];
    const int*   src   = (const int*)  d_in[1];
    const int*   dst   = (const int*)  d_in[2];
    const float* W_in  = (const float*)d_in[3];
    const float* b_in  = (const float*)d_in[4];
    const float* W_lin = (const float*)d_in[5];
    const float* b_lin = (const float*)d_in[6];
    const float* eps   = (const float*)d_in[7];
    const float* W_out = (const float*)d_in[8];
    const float* b_out = (const float*)d_in[9];

    const int M = in_sizes[0] / KDIM;   // 100000 (IN_DIM == HID == 128)
    const int E = in_sizes[1];          // 1600000

    // Workspace carve-out: h (51.2MB) | z (51.2MB) | cval (6.4MB) | cidx (6.4MB)
    float* hbuf = (float*)d_ws;
    float* zbuf = hbuf + (size_t)M * HID;
    float* cval = zbuf + (size_t)M * HID;
    int*   cidx = (int*)(cval + (size_t)M * MAXKK);

    const int strips     = M / 16;                 // 6250, exact
    const int gemmBlocks = (strips + 7) / 8;       // 8 waves/block
    const int rowBlocks  = (M + 7) / 8;            // maxk: 8 rows/block
    const long long eTh  = (long long)E * MAXKK;   // 25.6M lanes
    const int edgeBlocks = (int)((eTh + 255) / 256);

    // h = relu(x @ W_in + b_in)
    gemm_bias_kernel<HID, true><<<gemmBlocks, 256, 0, stream>>>(x, W_in, b_in, hbuf, M);

    for (int layer = 0; layer < 2; ++layer) {
        // z = h @ W_lin[layer] + b_lin[layer]
        gemm_bias_kernel<HID, false><<<gemmBlocks, 256, 0, stream>>>(
            hbuf, W_lin + (size_t)layer * HID * HID, b_lin + (size_t)layer * HID, zbuf, M);
        // h = (1+eps)*maxk(z); emit compact (val,idx)
        maxk_eps_kernel<<<rowBlocks, 256, 0, stream>>>(zbuf, eps, layer, hbuf, cval, cidx, M);
        // h[dst] += maxk(z)[src]
        edge_scatter_kernel<<<edgeBlocks, 256, 0, stream>>>(src, dst, cval, cidx, hbuf, E);
    }

    // out = h @ W_out + b_out
    gemm_bias_kernel<64, false><<<gemmBlocks, 256, 0, stream>>>(
        hbuf, W_out, b_out, (float*)d_out, M);
}